// ListwiseSmoothINDCGKLoss_31164282700344
// MI455X (gfx1250) — compile-verified
//
#include <hip/hip_runtime.h>
#include <hip/hip_bf16.h>
#include <stdint.h>

#define LROW  1024
#define BLK   128           /* 4 wave32 = 4 independent rows per block */
#define ROWS_PER_BLK 4
#define EPT   32            /* elements per lane: 32 lanes * 32 = 1024 */
#define KTOP  10
#define EPSF  2.220446049250313e-16f
#define AL2E  14.426950408889634f   /* ALPHA * log2(e) */
#define OMD   0.9f                  /* 1 - DELTA */

#if defined(__has_builtin)
#if __has_builtin(__builtin_amdgcn_global_load_async_to_lds_b128)
#define HAVE_ASYNC_LDS 1
#endif
#if __has_builtin(__builtin_amdgcn_mov_dpp8) && \
    __has_builtin(__builtin_amdgcn_update_dpp) && \
    __has_builtin(__builtin_amdgcn_permlanex16)
#define HAVE_FAST_BFLY 1
#endif
#endif
#ifndef HAVE_ASYNC_LDS
#define HAVE_ASYNC_LDS 0
#endif
#ifndef HAVE_FAST_BFLY
#define HAVE_FAST_BFLY 0
#endif

typedef int v4i __attribute__((vector_size(16)));
typedef __attribute__((address_space(1))) v4i* gv4i_p;
typedef __attribute__((address_space(3))) v4i* lv4i_p;

// 1/log2(k+2), k=0..9 (device copy for the dynamic IDCG walk)
static __device__ const float INV_DENOM_G[KTOP] = {
    1.0f,           0.63092975357f, 0.5f,           0.43067655808f,
    0.38685280724f, 0.35620718711f, 0.33333333333f, 0.31546487678f,
    0.30102999566f, 0.28906482631f};

// ---- wave32 butterfly reduction primitives (pure VALU, no LDS) ----
#define DPP8_SEL(a,b,c,d,e,f,g,h) \
  ((a)|((b)<<3)|((c)<<6)|((d)<<9)|((e)<<12)|((f)<<15)|((g)<<18)|((h)<<21))
#define DPP8_XOR1 DPP8_SEL(1,0,3,2,5,4,7,6)
#define DPP8_XOR2 DPP8_SEL(2,3,0,1,6,7,4,5)
#define DPP8_XOR4 DPP8_SEL(4,5,6,7,0,1,2,3)
#define DPP_ROW_MIRROR 0x140

__device__ __forceinline__ float bfly_x1(float v) {
#if HAVE_FAST_BFLY
  return __int_as_float(__builtin_amdgcn_mov_dpp8(__float_as_int(v), DPP8_XOR1));
#else
  return __shfl_xor(v, 1, 32);
#endif
}
__device__ __forceinline__ float bfly_x2(float v) {
#if HAVE_FAST_BFLY
  return __int_as_float(__builtin_amdgcn_mov_dpp8(__float_as_int(v), DPP8_XOR2));
#else
  return __shfl_xor(v, 2, 32);
#endif
}
__device__ __forceinline__ float bfly_x4(float v) {
#if HAVE_FAST_BFLY
  return __int_as_float(__builtin_amdgcn_mov_dpp8(__float_as_int(v), DPP8_XOR4));
#else
  return __shfl_xor(v, 4, 32);
#endif
}
__device__ __forceinline__ float bfly_x8(float v) {  // valid once 8-groups uniform
#if HAVE_FAST_BFLY
  return __int_as_float(__builtin_amdgcn_update_dpp(
      __float_as_int(v), __float_as_int(v), DPP_ROW_MIRROR, 0xF, 0xF, true));
#else
  return __shfl_xor(v, 8, 32);
#endif
}
__device__ __forceinline__ float bfly_x16(float v) {
#if HAVE_FAST_BFLY
  return __int_as_float(__builtin_amdgcn_permlanex16(
      __float_as_int(v), __float_as_int(v), 0x76543210, (int)0xFEDCBA98,
      false, false));
#else
  return __shfl_xor(v, 16, 32);
#endif
}

__device__ __forceinline__ float wred_sum(float v) {
  v += bfly_x1(v); v += bfly_x2(v); v += bfly_x4(v);
  v += bfly_x8(v); v += bfly_x16(v);
  return v;
}
__device__ __forceinline__ float wred_max(float v) {
  v = fmaxf(v, bfly_x1(v)); v = fmaxf(v, bfly_x2(v)); v = fmaxf(v, bfly_x4(v));
  v = fmaxf(v, bfly_x8(v)); v = fmaxf(v, bfly_x16(v));
  return v;
}
__device__ __forceinline__ float wred_min(float v) {
  v = fminf(v, bfly_x1(v)); v = fminf(v, bfly_x2(v)); v = fminf(v, bfly_x4(v));
  v = fminf(v, bfly_x8(v)); v = fminf(v, bfly_x16(v));
  return v;
}
__device__ __forceinline__ int wred_sum_i(int v) {  // packed 2x16-bit counters
#if HAVE_FAST_BFLY
  v += __builtin_amdgcn_mov_dpp8(v, DPP8_XOR1);
  v += __builtin_amdgcn_mov_dpp8(v, DPP8_XOR2);
  v += __builtin_amdgcn_mov_dpp8(v, DPP8_XOR4);
  v += __builtin_amdgcn_update_dpp(v, v, DPP_ROW_MIRROR, 0xF, 0xF, true);
  v += __builtin_amdgcn_permlanex16(v, v, 0x76543210, (int)0xFEDCBA98, false,
                                    false);
#else
  v += __shfl_xor(v, 1, 32);  v += __shfl_xor(v, 2, 32);
  v += __shfl_xor(v, 4, 32);  v += __shfl_xor(v, 8, 32);
  v += __shfl_xor(v, 16, 32);
#endif
  return v;
}

__global__ __launch_bounds__(BLK) void ndcg_row_kernel(
    const float* __restrict__ s, const int* __restrict__ label,
    float* __restrict__ row_loss) {
  const int lane = threadIdx.x & 31;
  const int wid  = threadIdx.x >> 5;
  const int row  = blockIdx.x * ROWS_PER_BLK + wid;

  __shared__ int lds_lab[ROWS_PER_BLK][LROW];

  const float* __restrict__ srow = s + (size_t)row * LROW;
  const int*   __restrict__ lrow = label + (size_t)row * LROW;

  // CDNA5 async DMA: stage this wave's labels into its LDS slice (8xB128/lane),
  // overlapped with the s load + min reduction below. Tracked by ASYNCcnt.
#if HAVE_ASYNC_LDS
#pragma unroll
  for (int j = 0; j < 8; ++j) {
    const int q = j * 32 + lane;  // int4 index
    __builtin_amdgcn_global_load_async_to_lds_b128(
        (gv4i_p)(void*)(const_cast<int*>(lrow) + 4 * q),
        (lv4i_p)(void*)&lds_lab[wid][4 * q], 0, 0);
  }
#endif

  // ---- load s (coalesced B128), row min ----
  float sp[EPT];
  float mn = 3.402823466e38f;
#pragma unroll
  for (int j = 0; j < 8; ++j) {
    const float4 v = ((const float4*)srow)[j * 32 + lane];
    sp[4 * j + 0] = v.x; sp[4 * j + 1] = v.y;
    sp[4 * j + 2] = v.z; sp[4 * j + 3] = v.w;
    mn = fminf(mn, fminf(fminf(v.x, v.y), fminf(v.z, v.w)));
  }
  const float rmin = wred_min(mn);

#pragma unroll
  for (int i = 0; i < EPT; ++i) sp[i] = (sp[i] - rmin) * AL2E;

  // ---- labels (via async LDS staging), packed histogram counters ----
  float labf[EPT];
  int pA = 0, pB = 0;  // pA = c1 | c2<<16 ; pB = c3 | c4<<16
#if HAVE_ASYNC_LDS
  asm volatile("s_wait_asynccnt 0x0" ::: "memory");  // wave reads its own data
#endif
#pragma unroll
  for (int j = 0; j < 8; ++j) {
    const int q = j * 32 + lane;
#if HAVE_ASYNC_LDS
    const int4 li = ((const int4*)lds_lab[wid])[q];
#else
    const int4 li = ((const int4*)lrow)[q];
#endif
    const int x0 = li.x, x1 = li.y, x2 = li.z, x3 = li.w;
    labf[4 * j + 0] = (float)x0; labf[4 * j + 1] = (float)x1;
    labf[4 * j + 2] = (float)x2; labf[4 * j + 3] = (float)x3;
    pA += (x0 == 1) + (x1 == 1) + (x2 == 1) + (x3 == 1);
    pA += ((x0 == 2) + (x1 == 2) + (x2 == 2) + (x3 == 2)) << 16;
    pB += (x0 == 3) + (x1 == 3) + (x2 == 3) + (x3 == 3);
    pB += ((x0 == 4) + (x1 == 4) + (x2 == 4) + (x3 == 4)) << 16;
  }
  pA = wred_sum_i(pA);
  pB = wred_sum_i(pB);

  // ---- K smooth-softmax iterations (all wave-local) ----
  constexpr float INVD[KTOP] = {
      1.0f,           0.63092975357f, 0.5f,           0.43067655808f,
      0.38685280724f, 0.35620718711f, 0.33333333333f, 0.31546487678f,
      0.30102999566f, 0.28906482631f};

  float prod[EPT], e[EPT];
#pragma unroll
  for (int i = 0; i < EPT; ++i) prod[i] = 1.f;
  float dcg = 0.f;

#pragma unroll
  for (int k = 0; k < KTOP; ++k) {
    float bm = -3.402823466e38f;
#pragma unroll
    for (int i = 0; i < EPT; ++i) {
      e[i] = sp[i] * prod[i];
      bm = fmaxf(bm, e[i]);
    }
    const float BM = wred_max(bm);

    float se = 0.f, sl = 0.f;
#pragma unroll
    for (int i = 0; i < EPT; ++i) {
      const float ev = __builtin_amdgcn_exp2f(e[i] - BM);
      e[i] = ev;
      se += ev;
      sl = fmaf(labf[i], ev, sl);
    }
    se = wred_sum(se);
    sl = wred_sum(sl);
    const float inv = __builtin_amdgcn_rcpf(se);

    dcg += __builtin_amdgcn_exp2f(sl * inv) * INVD[k];

#pragma unroll
    for (int i = 0; i < EPT; ++i) prod[i] *= fmaf(e[i], -inv, OMD);
  }

  // ---- ideal DCG from label histogram, final row loss ----
  if (lane == 0) {
    int cnt[5];
    cnt[1] = pA & 0xFFFF; cnt[2] = (pA >> 16) & 0xFFFF;
    cnt[3] = pB & 0xFFFF; cnt[4] = (pB >> 16) & 0xFFFF;
    cnt[0] = LROW - cnt[1] - cnt[2] - cnt[3] - cnt[4];
    float idcg = EPSF;
    int rem = KTOP, pos = 0;
    for (int v = 4; v >= 0 && rem > 0; --v) {
      int c = cnt[v];
      if (c > rem) c = rem;
      const float pw = (float)(1 << v);  // 2^label
      for (int j = 0; j < c; ++j) { idcg += pw * INV_DENOM_G[pos]; ++pos; }
      rem -= c;
    }
    row_loss[row] = 1.0f - (dcg + EPSF) * __builtin_amdgcn_rcpf(idcg);
  }
}

// Deterministic fixed-order tree reduction of per-row losses -> scalar.
__global__ __launch_bounds__(1024) void final_reduce(
    const float* __restrict__ v, float* __restrict__ out, int n) {
  __shared__ float sm[1024];
  float acc = 0.f;
  for (int i = threadIdx.x; i < n; i += 1024) acc += v[i];
  sm[threadIdx.x] = acc;
  __syncthreads();
#pragma unroll
  for (int s = 512; s > 0; s >>= 1) {
    if ((int)threadIdx.x < s) sm[threadIdx.x] += sm[threadIdx.x + s];
    __syncthreads();
  }
  if (threadIdx.x == 0) out[0] = sm[0];
}

extern "C" void kernel_launch(void* const* d_in, const int* in_sizes, int n_in,
                              void* d_out, int out_size, void* d_ws, size_t ws_size,
                              hipStream_t stream) {
  const float* s     = (const float*)d_in[0];
  const int*   label = (const int*)d_in[1];
  float*       out   = (float*)d_out;
  float*       rows  = (float*)d_ws;  // 32768 floats = 128 KiB scratch

  const int B = in_sizes[0] / LROW;
  ndcg_row_kernel<<<B / ROWS_PER_BLK, BLK, 0, stream>>>(s, label, rows);
  final_reduce<<<1, 1024, 0, stream>>>(rows, out, B);
}